// Model_39522289058322
// MI455X (gfx1250) — compile-verified
//
#include <hip/hip_runtime.h>
#include <hip/hip_bf16.h>
#include <stdint.h>

// ---------------------------------------------------------------------------
// GCN pipeline for MI455X (gfx1250, wave32).
// Memory-bound workload -> fp32 end to end; GEMMs use V_WMMA_F32_16X16X4_F32
// with B tiles staged through LDS via GLOBAL_LOAD_ASYNC_TO_LDS.
// ---------------------------------------------------------------------------

typedef float v2f __attribute__((ext_vector_type(2)));
typedef float v8f __attribute__((ext_vector_type(8)));

#define N_M   8000
#define N_D   8000
#define FGDIM 512
#define E_M   256000
#define E_D   256000
#define KC    32      // K-chunk staged per LDS buffer (k_gemm_wmma_lds)
#define KNT   64      // fixed K of the final x @ y^T GEMM

// ---------------------------------------------------------------------------
__global__ void k_fill(float* __restrict__ p, float v, long long n) {
    long long i = (long long)blockIdx.x * blockDim.x + threadIdx.x;
    long long stride = (long long)gridDim.x * blockDim.x;
    for (; i < n; i += stride) p[i] = v;
}

// w[e] = relu(dense[row[e], col[e]]);  deg[col[e]] += w[e]  (deg pre-filled 1.0)
__global__ void k_edge_weight(const int* __restrict__ row, const int* __restrict__ col,
                              const float* __restrict__ dense, float* __restrict__ w,
                              float* __restrict__ deg, int E, int N) {
    int e = blockIdx.x * blockDim.x + threadIdx.x;
    if (e >= E) return;
    int r = row[e], c = col[e];
    float v = dense[(long long)r * N + c];
    v = fmaxf(v, 0.0f);
    w[e] = v;
    atomicAdd(&deg[c], v);
}

// In-place: deg -> dinv = rsqrt(deg)
__global__ void k_dinv(float* __restrict__ deg, int n) {
    int i = blockIdx.x * blockDim.x + threadIdx.x;
    if (i >= n) return;
    float d = deg[i];
    deg[i] = (d > 0.0f) ? rsqrtf(d) : 0.0f;
}

// In-place: w[e] -> norm[e] = dinv[row]*w*dinv[col]
__global__ void k_norm(const int* __restrict__ row, const int* __restrict__ col,
                       const float* __restrict__ dinv, float* __restrict__ w, int E) {
    int e = blockIdx.x * blockDim.x + threadIdx.x;
    if (e >= E) return;
    w[e] = dinv[row[e]] * w[e] * dinv[col[e]];
}

// agg[col[e], f] += norm[e] * xw[row[e], f]   (F = 512 hardcoded)
__global__ void k_scatter(const int* __restrict__ row, const int* __restrict__ col,
                          const float* __restrict__ norm, const float* __restrict__ xw,
                          float* __restrict__ agg, long long total) {
    long long i = (long long)blockIdx.x * blockDim.x + threadIdx.x;
    long long stride = (long long)gridDim.x * blockDim.x;
    for (; i < total; i += stride) {
        long long e = i >> 9;
        int f = (int)(i & (FGDIM - 1));
        int r = row[e], c = col[e];
        float v = norm[e] * xw[(long long)r * FGDIM + f];
        atomicAdd(&agg[(long long)c * FGDIM + f], v);
    }
}

// out[i,f] = relu(agg[i,f] + dinv[i]^2 * xw[i,f] + bias[f])
__global__ void k_combine(const float* __restrict__ agg, const float* __restrict__ xw,
                          const float* __restrict__ dinv, const float* __restrict__ bias,
                          float* __restrict__ out, long long total) {
    long long i = (long long)blockIdx.x * blockDim.x + threadIdx.x;
    if (i >= total) return;
    long long node = i >> 9;
    int f = (int)(i & (FGDIM - 1));
    float di = dinv[node];
    float v = agg[i] + di * di * xw[i] + bias[f];
    out[i] = fmaxf(v, 0.0f);
}

// ---------------------------------------------------------------------------
// WMMA fp32 GEMM (row-major B), B tile staged via async-to-LDS, double buffer.
//   block = 128 threads = 4 waves; wave w owns rows [by*64 + w*16, +16),
//   block owns cols [bx*64, +64). Each wave: 4 accumulators (16x64 strip).
//   Requires M%64==0, N%64==0, K%KC==0 (all shapes here comply -> EXEC all-1s).
// ---------------------------------------------------------------------------
template <bool BIAS, bool RELU>
__global__ __launch_bounds__(128)
void k_gemm_wmma_lds(const float* __restrict__ A, const float* __restrict__ B,
                     const float* __restrict__ bias, float* __restrict__ C,
                     int N, int K) {
    __shared__ float lb[2][KC * 64];

    const int tid   = threadIdx.x;
    const int lane  = tid & 31;
    const int wave  = tid >> 5;
    const int row0  = blockIdx.y * 64 + wave * 16;
    const int n0    = blockIdx.x * 64;
    const int mrow  = lane & 15;
    const int khalf = lane >> 4;       // 0 -> K{0,1}, 1 -> K{2,3}

    v8f acc[4] = {v8f{}, v8f{}, v8f{}, v8f{}};

    // Stage one KC x 64 tile of B into lb[buf]: 512 float4s over 128 threads,
    // 4 async b128 instructions per wave (ASYNCcnt += 4 per wave per stage).
    auto stage = [&](int buf, int k0) {
#pragma unroll
        for (int j = 0; j < 4; ++j) {
            const int idx4 = tid + j * 128;        // [0,512)
            const int krow = idx4 >> 4;            // [0,32)
            const int c4   = (idx4 & 15) << 2;     // [0,64) step 4
            const float* gp = B + (long long)(k0 + krow) * N + n0 + c4;
            const uint32_t lp = (uint32_t)(uintptr_t)&lb[buf][krow * 64 + c4];
            asm volatile("global_load_async_to_lds_b128 %0, %1, off"
                         :: "v"(lp), "v"(gp) : "memory");
        }
    };

    stage(0, 0);
    const int NCH = K / KC;
    const float* Arow = A + (long long)(row0 + mrow) * K + 2 * khalf;

    for (int ch = 0; ch < NCH; ++ch) {
        const int cur = ch & 1;
        const int k0  = ch * KC;
        if (ch + 1 < NCH) {
            stage(cur ^ 1, k0 + KC);
            // in-order: <=4 outstanding means chunk `ch` is fully in LDS,
            // chunk `ch+1` stays in flight behind the compute below.
            asm volatile("s_wait_asynccnt 0x4" ::: "memory");
        } else {
            asm volatile("s_wait_asynccnt 0x0" ::: "memory");
        }
        __syncthreads();

        // Preload this chunk's A fragments (contiguous b64 per k-step).
        v2f afr[KC / 4];
#pragma unroll
        for (int s = 0; s < KC / 4; ++s)
            afr[s] = *(const v2f*)(Arow + k0 + s * 4);

        const float* lbp = lb[cur];
#pragma unroll
        for (int s = 0; s < KC / 4; ++s) {
            const int kk = s * 4 + 2 * khalf;
#pragma unroll
            for (int nt = 0; nt < 4; ++nt) {
                const int c = nt * 16 + mrow;
                v2f b;
                b.x = lbp[kk * 64 + c];
                b.y = lbp[(kk + 1) * 64 + c];
                acc[nt] = __builtin_amdgcn_wmma_f32_16x16x4_f32(
                    false, afr[s], false, b, (short)0, acc[nt], false, false);
            }
        }
        __syncthreads();   // all waves done reading lb[cur] before it is restaged
    }

#pragma unroll
    for (int nt = 0; nt < 4; ++nt) {
        const int n = n0 + nt * 16 + mrow;
        const float bv = BIAS ? bias[n] : 0.0f;
#pragma unroll
        for (int j = 0; j < 8; ++j) {
            const int m = row0 + j + 8 * khalf;
            float v = acc[nt][j] + bv;
            if (RELU) v = fmaxf(v, 0.0f);
            C[(long long)m * N + n] = v;
        }
    }
}

// ---------------------------------------------------------------------------
// WMMA fp32 GEMM, C = A[M,KNT] @ B^T where B is [N,KNT] row-major (x @ y^T).
// K fixed at 64: the block's whole 64x64 B tile (16 KB) is staged once into
// LDS via async-to-LDS; all A fragments preloaded -> hot loop is ds+wmma only.
// ---------------------------------------------------------------------------
__global__ __launch_bounds__(128)
void k_gemm_wmma_nt(const float* __restrict__ A, const float* __restrict__ B,
                    float* __restrict__ C, int N) {
    __shared__ float lb[64 * KNT];     // lb[col_local * 64 + k]

    const int tid   = threadIdx.x;
    const int lane  = tid & 31;
    const int wave  = tid >> 5;
    const int row0  = blockIdx.y * 64 + wave * 16;
    const int n0    = blockIdx.x * 64;
    const int mrow  = lane & 15;
    const int khalf = lane >> 4;

    // Stage B tile: rows yf[n0 + r][0..63] -> lb[r*64 ..]; 1024 float4s over
    // 128 threads = 8 async b128 instructions per wave.
#pragma unroll
    for (int j = 0; j < 8; ++j) {
        const int idx4 = tid + j * 128;        // [0,1024)
        const int r    = idx4 >> 4;            // [0,64)
        const int c4   = (idx4 & 15) << 2;     // [0,64) step 4
        const float* gp = B + (long long)(n0 + r) * KNT + c4;
        const uint32_t lp = (uint32_t)(uintptr_t)&lb[r * KNT + c4];
        asm volatile("global_load_async_to_lds_b128 %0, %1, off"
                     :: "v"(lp), "v"(gp) : "memory");
    }

    // Preload all A fragments for K=64 while the async tile is in flight.
    const float* Arow = A + (long long)(row0 + mrow) * KNT + 2 * khalf;
    v2f afr[KNT / 4];
#pragma unroll
    for (int s = 0; s < KNT / 4; ++s)
        afr[s] = *(const v2f*)(Arow + s * 4);

    asm volatile("s_wait_asynccnt 0x0" ::: "memory");
    __syncthreads();

    v8f acc[4] = {v8f{}, v8f{}, v8f{}, v8f{}};
#pragma unroll
    for (int s = 0; s < KNT / 4; ++s) {
        const int kk = s * 4 + 2 * khalf;
#pragma unroll
        for (int nt = 0; nt < 4; ++nt) {
            const int c = nt * 16 + mrow;
            v2f b = *(const v2f*)&lb[c * KNT + kk];   // aligned: kk even
            acc[nt] = __builtin_amdgcn_wmma_f32_16x16x4_f32(
                false, afr[s], false, b, (short)0, acc[nt], false, false);
        }
    }

#pragma unroll
    for (int nt = 0; nt < 4; ++nt) {
        const int n = n0 + nt * 16 + mrow;
#pragma unroll
        for (int j = 0; j < 8; ++j) {
            const int m = row0 + j + 8 * khalf;
            C[(long long)m * N + n] = acc[nt][j];
        }
    }
}

// ---------------------------------------------------------------------------
extern "C" void kernel_launch(void* const* d_in, const int* in_sizes, int n_in,
                              void* d_out, int out_size, void* d_ws, size_t ws_size,
                              hipStream_t stream) {
    (void)in_sizes; (void)n_in; (void)out_size; (void)ws_size;

    const float* x_m    = (const float*)d_in[0];
    const float* x_d    = (const float*)d_in[1];
    const float* data_m = (const float*)d_in[2];
    const float* data_d = (const float*)d_in[3];
    const int*   ei_m   = (const int*)d_in[4];
    const int*   ei_d   = (const int*)d_in[5];
    const float* Wg1 = (const float*)d_in[6];  const float* bg1 = (const float*)d_in[7];
    const float* Wg2 = (const float*)d_in[8];  const float* bg2 = (const float*)d_in[9];
    const float* Wd1 = (const float*)d_in[10]; const float* bd1 = (const float*)d_in[11];
    const float* Wd2 = (const float*)d_in[12]; const float* bd2 = (const float*)d_in[13];
    const float* lx1w = (const float*)d_in[14]; const float* lx1b = (const float*)d_in[15];
    const float* lx2w = (const float*)d_in[16]; const float* lx2b = (const float*)d_in[17];
    const float* lx3w = (const float*)d_in[18]; const float* lx3b = (const float*)d_in[19];
    const float* ly1w = (const float*)d_in[20]; const float* ly1b = (const float*)d_in[21];
    const float* ly2w = (const float*)d_in[22]; const float* ly2b = (const float*)d_in[23];
    const float* ly3w = (const float*)d_in[24]; const float* ly3b = (const float*)d_in[25];
    float* out = (float*)d_out;

    // ---- workspace carve-out (floats) ----
    float* ws = (float*)d_ws;
    size_t off = 0;
    auto carve = [&](size_t n) { float* p = ws + off; off += n; return p; };
    float* norm_m = carve(E_M);
    float* norm_d = carve(E_D);
    float* dinv_m = carve(N_M);
    float* dinv_d = carve(N_D);
    float* xw     = carve((size_t)N_M * FGDIM);
    float* agg    = carve((size_t)N_M * FGDIM);
    float* X1     = carve((size_t)N_M * FGDIM);
    float* Xg     = carve((size_t)N_M * FGDIM);
    float* Y1     = carve((size_t)N_D * FGDIM);
    float* Yg     = carve((size_t)N_D * FGDIM);
    float* t1     = carve((size_t)N_M * 256);
    float* t2     = carve((size_t)N_M * 128);
    float* xf     = carve((size_t)N_M * 64);
    float* yf     = carve((size_t)N_D * 64);

    const int* row_m = ei_m;            const int* col_m = ei_m + E_M;
    const int* row_d = ei_d;            const int* col_d = ei_d + E_D;

    // ---- edge weights, degrees, norms ----
    k_fill<<<64, 256, 0, stream>>>(dinv_m, 1.0f, N_M);
    k_fill<<<64, 256, 0, stream>>>(dinv_d, 1.0f, N_D);
    k_edge_weight<<<(E_M + 255) / 256, 256, 0, stream>>>(row_m, col_m, data_m, norm_m, dinv_m, E_M, N_M);
    k_edge_weight<<<(E_D + 255) / 256, 256, 0, stream>>>(row_d, col_d, data_d, norm_d, dinv_d, E_D, N_D);
    k_dinv<<<(N_M + 255) / 256, 256, 0, stream>>>(dinv_m, N_M);
    k_dinv<<<(N_D + 255) / 256, 256, 0, stream>>>(dinv_d, N_D);
    k_norm<<<(E_M + 255) / 256, 256, 0, stream>>>(row_m, col_m, dinv_m, norm_m, E_M);
    k_norm<<<(E_D + 255) / 256, 256, 0, stream>>>(row_d, col_d, dinv_d, norm_d, E_D);

    // ---- one GCN conv layer ----
    auto gcn_layer = [&](const float* X, const float* W, const float* bias,
                         const int* row, const int* col, const float* norm,
                         const float* dinv, int E, float* Xout) {
        dim3 g(FGDIM / 64, N_M / 64);
        k_gemm_wmma_lds<false, false><<<g, 128, 0, stream>>>(X, W, nullptr, xw, FGDIM, FGDIM);
        const long long ntot = (long long)N_M * FGDIM;
        k_fill<<<4096, 256, 0, stream>>>(agg, 0.0f, ntot);
        const long long etot = (long long)E * FGDIM;
        k_scatter<<<(unsigned)((etot + 255) / 256), 256, 0, stream>>>(row, col, norm, xw, agg, etot);
        k_combine<<<(unsigned)((ntot + 255) / 256), 256, 0, stream>>>(agg, xw, dinv, bias, Xout, ntot);
    };

    gcn_layer(x_m, Wg1, bg1, row_m, col_m, norm_m, dinv_m, E_M, X1);
    gcn_layer(X1,  Wg2, bg2, row_m, col_m, norm_m, dinv_m, E_M, Xg);
    gcn_layer(x_d, Wd1, bd1, row_d, col_d, norm_d, dinv_d, E_D, Y1);
    gcn_layer(Y1,  Wd2, bd2, row_d, col_d, norm_d, dinv_d, E_D, Yg);

    // ---- MLP projections (bias + relu fused) ----
    k_gemm_wmma_lds<true, true><<<dim3(256 / 64, N_M / 64), 128, 0, stream>>>(Xg, lx1w, lx1b, t1, 256, 512);
    k_gemm_wmma_lds<true, true><<<dim3(128 / 64, N_M / 64), 128, 0, stream>>>(t1, lx2w, lx2b, t2, 128, 256);
    k_gemm_wmma_lds<true, true><<<dim3(64  / 64, N_M / 64), 128, 0, stream>>>(t2, lx3w, lx3b, xf, 64, 128);

    k_gemm_wmma_lds<true, true><<<dim3(256 / 64, N_D / 64), 128, 0, stream>>>(Yg, ly1w, ly1b, t1, 256, 512);
    k_gemm_wmma_lds<true, true><<<dim3(128 / 64, N_D / 64), 128, 0, stream>>>(t1, ly2w, ly2b, t2, 128, 256);
    k_gemm_wmma_lds<true, true><<<dim3(64  / 64, N_D / 64), 128, 0, stream>>>(t2, ly3w, ly3b, yf, 64, 128);

    // ---- pairwise scores: out = xf @ yf^T  [8000 x 8000], K = 64 ----
    k_gemm_wmma_nt<<<dim3(N_D / 64, N_M / 64), 128, 0, stream>>>(xf, yf, out, N_D);
}